// LocalAttention_57913339019897
// MI455X (gfx1250) — compile-verified
//
#include <hip/hip_runtime.h>
#include <hip/hip_bf16.h>
#include <stdint.h>

// Problem constants (from reference)
#define B_  2
#define T_  2048
#define D_  1024
#define H_  16
#define HD_ 64
#define WIN_ 512

typedef float v2f __attribute__((ext_vector_type(2)));
typedef float v8f __attribute__((ext_vector_type(8)));
typedef int   v4i __attribute__((vector_size(16)));

// ---------------------------------------------------------------------------
// Async global->LDS copy (CDNA5 GLOBAL_LOAD_ASYNC_TO_LDS_B128, ASYNCcnt).
// Probe-learned signature: (v4i AS1* src, v4i AS3* dst, imm offset, imm cpol).
// ---------------------------------------------------------------------------
#if __has_builtin(__builtin_amdgcn_global_load_async_to_lds_b128)
#define HAS_ASYNC_LDS 1
#else
#define HAS_ASYNC_LDS 0
#endif

__device__ __forceinline__ void async_cp16(const float* g, float* l)
{
#if HAS_ASYNC_LDS
    __builtin_amdgcn_global_load_async_to_lds_b128(
        (__attribute__((address_space(1))) v4i*)(uintptr_t)g,
        (__attribute__((address_space(3))) v4i*)(uint32_t)(uintptr_t)l,
        0, 0);
#else
    *(float4*)l = *(const float4*)g;
#endif
}

__device__ __forceinline__ void wait_async0()
{
#if HAS_ASYNC_LDS
#if __has_builtin(__builtin_amdgcn_s_wait_asynccnt)
    __builtin_amdgcn_s_wait_asynccnt(0);
#else
    asm volatile("s_wait_asynccnt 0x0" ::: "memory");
#endif
#endif
}

// ---------------------------------------------------------------------------
// GEMM: C(MxN) = A(MxK) * W(KxN), all row-major fp32.
// Block = 128 threads (4 waves). Workgroup tile 64x64, K staged 16 at a time
// through double-buffered LDS filled by async-to-LDS DMA; each wave computes
// a 16x64 strip = 4 WMMA f32 16x16 tiles (256 v_wmma per K-pass).
// ---------------------------------------------------------------------------
__global__ __launch_bounds__(128) void gemm_f32_wmma(
    const float* __restrict__ A, const float* __restrict__ W,
    float* __restrict__ C, int M, int N, int K)
{
    __shared__ __attribute__((aligned(16))) float As[2][64 * 16];
    __shared__ __attribute__((aligned(16))) float Ws[2][16 * 64];

    const int tid  = threadIdx.x;
    const int wave = tid >> 5;
    const int lane = tid & 31;
    const int half = lane >> 4;   // 0: lanes 0-15, 1: lanes 16-31
    const int ln   = lane & 15;

    const int rowBase = blockIdx.y * 64;
    const int colBase = blockIdx.x * 64;

    // staging assignment: 8 floats / thread for each of As (64x16), Ws (16x64)
    const int ar = tid >> 1, ac = (tid & 1) * 8;
    const int wr = tid >> 3, wc = (tid & 7) * 8;
    const float* aSrc = A + (size_t)(rowBase + ar) * K + ac;
    const float* wSrc = W + (size_t)wr * N + colBase + wc;

    auto issue_stage = [&](int buf, int k0) {
        const float* a0 = aSrc + k0;
        async_cp16(a0,     &As[buf][ar * 16 + ac]);
        async_cp16(a0 + 4, &As[buf][ar * 16 + ac + 4]);
        const float* w0 = wSrc + (size_t)k0 * N;
        async_cp16(w0,     &Ws[buf][wr * 64 + wc]);
        async_cp16(w0 + 4, &Ws[buf][wr * 64 + wc + 4]);
    };

    v8f acc[4];
    for (int t = 0; t < 4; ++t)
        for (int j = 0; j < 8; ++j) acc[t][j] = 0.0f;

    const int nStage = K / 16;
    issue_stage(0, 0);

    for (int s = 0; s < nStage; ++s) {
        const int cur = s & 1;
        wait_async0();        // this wave's fills for stage s have landed
        __syncthreads();      // everyone's fills landed; buf cur^1 free to overwrite
        if (s + 1 < nStage)
            issue_stage(cur ^ 1, (s + 1) * 16);   // overlap with compute below

        const float* as = &As[cur][0];
        const float* ws = &Ws[cur][0];
        #pragma unroll
        for (int kk = 0; kk < 16; kk += 4) {
            // A fragment: element (M=ln, K=kk + j + 2*half), j=0,1 consecutive
            v2f af = *(const v2f*)&as[(wave * 16 + ln) * 16 + kk + 2 * half];
            #pragma unroll
            for (int t = 0; t < 4; ++t) {
                // B fragment: element (K=kk + j + 2*half, N=t*16+ln)
                v2f bf;
                bf[0] = ws[(kk + 2 * half + 0) * 64 + t * 16 + ln];
                bf[1] = ws[(kk + 2 * half + 1) * 64 + t * 16 + ln];
                acc[t] = __builtin_amdgcn_wmma_f32_16x16x4_f32(
                    false, af, false, bf, (short)0, acc[t], false, false);
            }
        }
        __syncthreads();      // all waves done reading buf cur before refill
    }

    // C layout: vgpr j of lane -> (M = j + 8*half, N = ln)
    for (int t = 0; t < 4; ++t)
        for (int j = 0; j < 8; ++j)
            C[(size_t)(rowBase + wave * 16 + j + 8 * half) * N +
              colBase + t * 16 + ln] = acc[t][j];
}

// ---------------------------------------------------------------------------
// RMSNorm (over head dim 64) + RoPE on q,k; transpose q,k,v from
// (B,T,H,HD) rows to (B,H,T,HD). One wave per (b,t,h).
// ---------------------------------------------------------------------------
__global__ __launch_bounds__(128) void norm_rope_transpose(
    const float* __restrict__ qr, const float* __restrict__ kr,
    const float* __restrict__ vr, const float* __restrict__ qs,
    const float* __restrict__ ks, float* __restrict__ qt,
    float* __restrict__ kt, float* __restrict__ vt)
{
    const int gw   = blockIdx.x * 4 + (threadIdx.x >> 5); // wave id over B*T*H
    const int lane = threadIdx.x & 31;
    const int h = gw % H_;
    const int t = (gw / H_) % T_;
    const int b = gw / (H_ * T_);

    const size_t inOff  = ((size_t)(b * T_ + t)) * D_ + h * HD_;
    const size_t outOff = (((size_t)b * H_ + h) * T_ + t) * HD_;

    // RoPE angle for dim pair (lane, lane+32): inv_freq = base^(-2*lane/64)
    float invf = __powf(10000.0f, -(2.0f * (float)lane) / 64.0f);
    float ang  = (float)t * invf;
    float c, s;
    __sincosf(ang, &s, &c);

    // ---- q ----
    {
        float e0 = qr[inOff + lane];
        float e1 = qr[inOff + lane + 32];
        float ssq = e0 * e0 + e1 * e1;
        for (int off = 16; off >= 1; off >>= 1)
            ssq += __shfl_xor(ssq, off, 32);
        float r = rsqrtf(ssq * (1.0f / 64.0f) + 1e-6f);
        e0 *= r * qs[lane];
        e1 *= r * qs[lane + 32];
        qt[outOff + lane]      = e0 * c - e1 * s;
        qt[outOff + lane + 32] = e1 * c + e0 * s;
    }
    // ---- k ----
    {
        float e0 = kr[inOff + lane];
        float e1 = kr[inOff + lane + 32];
        float ssq = e0 * e0 + e1 * e1;
        for (int off = 16; off >= 1; off >>= 1)
            ssq += __shfl_xor(ssq, off, 32);
        float r = rsqrtf(ssq * (1.0f / 64.0f) + 1e-6f);
        e0 *= r * ks[lane];
        e1 *= r * ks[lane + 32];
        kt[outOff + lane]      = e0 * c - e1 * s;
        kt[outOff + lane + 32] = e1 * c + e0 * s;
    }
    // ---- v (transpose only) ----
    vt[outOff + lane]      = vr[inOff + lane];
    vt[outOff + lane + 32] = vr[inOff + lane + 32];
}

// ---------------------------------------------------------------------------
// Flash-style windowed attention with sink logit.
// One wave per (b, h, 16-query tile). q/k/v in (B,H,T,64) layout.
// Output written in (B,T,H*HD) row-major so the final GEMM consumes it.
// ---------------------------------------------------------------------------
__global__ __launch_bounds__(32) void attn_wmma(
    const float* __restrict__ q, const float* __restrict__ k,
    const float* __restrict__ v, const float* __restrict__ sink,
    float* __restrict__ out)
{
    __shared__ __attribute__((aligned(16))) float Pl[16 * 16];

    const int lane = threadIdx.x & 31;
    const int half = lane >> 4;
    const int ln   = lane & 15;
    const int t0   = blockIdx.x * 16;
    const int h    = blockIdx.y;
    const int b    = blockIdx.z;

    const float* qb = q + (((size_t)b * H_ + h) * T_) * HD_;
    const float* kb = k + (((size_t)b * H_ + h) * T_) * HD_;
    const float* vb = v + (((size_t)b * H_ + h) * T_) * HD_;

    const float scale = 0.125f; // 64^-0.5
    const float NEG   = -1e30f;

    // Q fragments: A element (M=ln -> row t0+ln, K=d0*4 + j + 2*half)
    v2f qa[16];
    #pragma unroll
    for (int d0 = 0; d0 < 16; ++d0)
        qa[d0] = *(const v2f*)(qb + (size_t)(t0 + ln) * HD_ + d0 * 4 + 2 * half);

    v8f O[4];
    for (int t = 0; t < 4; ++t)
        for (int j = 0; j < 8; ++j) O[t][j] = 0.0f;
    float mrow[8], lrow[8];
    for (int j = 0; j < 8; ++j) { mrow[j] = NEG; lrow[j] = 0.0f; }

    int smin = t0 - (WIN_ - 1);
    if (smin < 0) smin = 0;
    const int s0start = smin & ~15;

    for (int s0 = s0start; s0 <= t0; s0 += 16) {
        // ---- scores S = Q * K^T (16x16, K-dim = 64) ----
        v8f S;
        for (int j = 0; j < 8; ++j) S[j] = 0.0f;
        #pragma unroll
        for (int d0 = 0; d0 < 16; ++d0) {
            // B element (K=d0*4 + j + 2*half, N=ln) = K[s0+ln][d...]
            v2f kf = *(const v2f*)(kb + (size_t)(s0 + ln) * HD_ + d0 * 4 + 2 * half);
            S = __builtin_amdgcn_wmma_f32_16x16x4_f32(
                false, qa[d0], false, kf, (short)0, S, false, false);
        }

        // ---- mask + online softmax ----
        float alpha[8];
        #pragma unroll
        for (int j = 0; j < 8; ++j) {
            int m = j + 8 * half;
            int t = t0 + m;
            int s = s0 + ln;
            float sc = S[j] * scale;
            bool valid = (s <= t) && ((t - s) < WIN_);
            sc = valid ? sc : NEG;
            float rm = sc;
            rm = fmaxf(rm, __shfl_xor(rm, 1, 32));
            rm = fmaxf(rm, __shfl_xor(rm, 2, 32));
            rm = fmaxf(rm, __shfl_xor(rm, 4, 32));
            rm = fmaxf(rm, __shfl_xor(rm, 8, 32));
            float mnew = fmaxf(mrow[j], rm);
            float al   = __expf(mrow[j] - mnew);
            float pe   = __expf(sc - mnew);
            float rs = pe;
            rs += __shfl_xor(rs, 1, 32);
            rs += __shfl_xor(rs, 2, 32);
            rs += __shfl_xor(rs, 4, 32);
            rs += __shfl_xor(rs, 8, 32);
            lrow[j]  = lrow[j] * al + rs;
            mrow[j]  = mnew;
            alpha[j] = al;
            Pl[m * 16 + ln] = pe;       // P in row-major for A-frag reload
        }
        // rescale running O
        for (int t = 0; t < 4; ++t)
            for (int j = 0; j < 8; ++j) O[t][j] *= alpha[j];

        __syncthreads();

        // ---- O += P * V (K-dim = 16) ----
        v2f pf[4];
        #pragma unroll
        for (int ss = 0; ss < 4; ++ss)
            pf[ss] = *(const v2f*)&Pl[ln * 16 + ss * 4 + 2 * half];
        #pragma unroll
        for (int c = 0; c < 4; ++c) {
            #pragma unroll
            for (int ss = 0; ss < 4; ++ss) {
                const float* vp =
                    vb + (size_t)(s0 + ss * 4 + 2 * half) * HD_ + c * 16 + ln;
                v2f vf;
                vf[0] = vp[0];
                vf[1] = vp[HD_];
                O[c] = __builtin_amdgcn_wmma_f32_16x16x4_f32(
                    false, pf[ss], false, vf, (short)0, O[c], false, false);
            }
        }
        __syncthreads();
    }

    // ---- fold in sink logit, normalize, store ----
    const float sl = sink[h];
    float rinv[8];
    #pragma unroll
    for (int j = 0; j < 8; ++j) {
        float mnew = fmaxf(mrow[j], sl);
        float al   = __expf(mrow[j] - mnew);
        float lfin = lrow[j] * al + __expf(sl - mnew);
        rinv[j] = al / lfin;
    }
    for (int c = 0; c < 4; ++c)
        for (int j = 0; j < 8; ++j) {
            int t = t0 + j + 8 * half;
            out[(size_t)(b * T_ + t) * D_ + h * HD_ + c * 16 + ln] =
                O[c][j] * rinv[j];
        }
}

// ---------------------------------------------------------------------------
extern "C" void kernel_launch(void* const* d_in, const int* in_sizes, int n_in,
                              void* d_out, int out_size, void* d_ws, size_t ws_size,
                              hipStream_t stream)
{
    const float* x    = (const float*)d_in[0];
    const float* Wq   = (const float*)d_in[1];
    const float* Wk   = (const float*)d_in[2];
    const float* Wv   = (const float*)d_in[3];
    const float* Wo   = (const float*)d_in[4];
    const float* qsc  = (const float*)d_in[5];
    const float* ksc  = (const float*)d_in[6];
    const float* sink = (const float*)d_in[7];
    float* out = (float*)d_out;

    const size_t CH = (size_t)B_ * T_ * D_; // 4M floats per buffer
    float* ws = (float*)d_ws;
    float* qr = ws + 0 * CH;
    float* kr = ws + 1 * CH;
    float* vr = ws + 2 * CH;
    float* qt = ws + 3 * CH;
    float* kt = ws + 4 * CH;
    float* vt = ws + 5 * CH;
    float* ao = ws + 6 * CH;

    const int M = B_ * T_, N = D_, K = D_;
    dim3 gg(N / 64, M / 64);

    gemm_f32_wmma<<<gg, 128, 0, stream>>>(x, Wq, qr, M, N, K);
    gemm_f32_wmma<<<gg, 128, 0, stream>>>(x, Wk, kr, M, N, K);
    gemm_f32_wmma<<<gg, 128, 0, stream>>>(x, Wv, vr, M, N, K);

    norm_rope_transpose<<<(B_ * T_ * H_) / 4, 128, 0, stream>>>(
        qr, kr, vr, qsc, ksc, qt, kt, vt);

    attn_wmma<<<dim3(T_ / 16, H_, B_), 32, 0, stream>>>(qt, kt, vt, sink, ao);

    gemm_f32_wmma<<<gg, 128, 0, stream>>>(ao, Wo, out, M, N, K);
}